// ColorLoss_7112465842712
// MI455X (gfx1250) — compile-verified
//
#include <hip/hip_runtime.h>
#include <hip/hip_bf16.h>

#define MAXBINS 4096   // supports n_bins up to 16 (16^3)

typedef __attribute__((ext_vector_type(2))) float v2f;
typedef __attribute__((ext_vector_type(8))) float v8f;

// ---------------------------------------------------------------------------
// Zero the histogram counter region of the workspace.
// ---------------------------------------------------------------------------
__global__ void zero_ws_kernel(unsigned* __restrict__ ws, int nwords) {
    int i = blockIdx.x * blockDim.x + threadIdx.x;
    if (i < nwords) ws[i] = 0u;
}

// searchsorted(boundaries, x, 'left') with boundaries i*(2/n)-1, i=1..n-1
// == clamp(floor((x+1)*n/2), 0, n-1) (exact off boundary points)
__device__ __forceinline__ int bin_idx(float x, float scale, float nbm1f) {
    float t = floorf(fmaf(x, scale, scale));   // (x + 1) * scale
    t = fminf(fmaxf(t, 0.0f), nbm1f);
    return (int)t;
}

// ---------------------------------------------------------------------------
// Per-image 3D color histogram with LDS privatization.
// grid = (blocksPerImage, B); img is [B,3,H,W]; hist is B x MAXBINS (u32).
// ---------------------------------------------------------------------------
__global__ __launch_bounds__(256) void hist3d_kernel(
    const float* __restrict__ img, unsigned* __restrict__ hist,
    const int* __restrict__ n_bins_ptr, int HW) {
    __shared__ unsigned lh[MAXBINS];

    const int nb   = n_bins_ptr[0];
    const int nb2  = nb * nb;
    const int ntot = nb2 * nb;
    const float scale = 0.5f * (float)nb;
    const float nbm1f = (float)(nb - 1);

    const int tid = threadIdx.x;
    for (int i = tid; i < MAXBINS; i += blockDim.x) lh[i] = 0u;
    __syncthreads();

    const size_t b = blockIdx.y;
    const float* pr = img + (b * 3 + 0) * (size_t)HW;
    const float* pg = img + (b * 3 + 1) * (size_t)HW;
    const float* pb = img + (b * 3 + 2) * (size_t)HW;
    unsigned* h = hist + b * (size_t)MAXBINS;

    const int nvec   = HW >> 2;                     // float4 chunks per plane
    const int stride = gridDim.x * blockDim.x;

    for (int v = blockIdx.x * blockDim.x + tid; v < nvec; v += stride) {
        // gfx1250 speculative prefetch for the next grid-stride iteration
        __builtin_prefetch(pr + (size_t)(v + stride) * 4, 0, 0);
        __builtin_prefetch(pg + (size_t)(v + stride) * 4, 0, 0);
        __builtin_prefetch(pb + (size_t)(v + stride) * 4, 0, 0);

        float4 r  = ((const float4*)pr)[v];         // global_load_b128
        float4 g  = ((const float4*)pg)[v];
        float4 bl = ((const float4*)pb)[v];

        int i0 = bin_idx(r.x, scale, nbm1f) + nb * bin_idx(g.x, scale, nbm1f)
               + nb2 * bin_idx(bl.x, scale, nbm1f);
        int i1 = bin_idx(r.y, scale, nbm1f) + nb * bin_idx(g.y, scale, nbm1f)
               + nb2 * bin_idx(bl.y, scale, nbm1f);
        int i2 = bin_idx(r.z, scale, nbm1f) + nb * bin_idx(g.z, scale, nbm1f)
               + nb2 * bin_idx(bl.z, scale, nbm1f);
        int i3 = bin_idx(r.w, scale, nbm1f) + nb * bin_idx(g.w, scale, nbm1f)
               + nb2 * bin_idx(bl.w, scale, nbm1f);

        atomicAdd(&lh[i0], 1u);                     // ds_add_u32
        atomicAdd(&lh[i1], 1u);
        atomicAdd(&lh[i2], 1u);
        atomicAdd(&lh[i3], 1u);
    }
    __syncthreads();

    // Sparse flush: Gaussian data leaves many of the 4096 bins empty per block
    for (int i = tid; i < ntot; i += blockDim.x) {
        unsigned v = lh[i];
        if (v) atomicAdd(&h[i], v);                 // integer atomics -> deterministic
    }
}

// ---------------------------------------------------------------------------
// Final L1-mean reduction, single 1024-thread block (deterministic).
// Intra-wave reduction uses V_WMMA_F32_16X16X4_F32 with B = ones:
//   D[m][n] = sum_k A[m][k]  ->  row sums; every column identical.
// Lane layout (wave32): A vgpr0 holds K=0 (lanes 0-15) / K=2 (lanes 16-31).
// Putting each lane's partial in a[0] and zero in a[1] makes
//   rowsum(m) = s_m + s_{m+16};  sum over the 8 D vgprs gives the half-wave
// sum in every lane; one shfl_xor(16) completes the 32-lane sum.
// ---------------------------------------------------------------------------
__global__ __launch_bounds__(1024) void loss_kernel(
    const unsigned* __restrict__ tgt, const unsigned* __restrict__ hin,
    const int* __restrict__ n_bins_ptr, float* __restrict__ out,
    int B, float invHW) {
    __shared__ float wsums[32];

    const int nb   = n_bins_ptr[0];
    const int ntot = nb * nb * nb;
    const int tid  = threadIdx.x;
    const int total = B * ntot;

    float acc = 0.0f;
    for (int i = tid; i < total; i += blockDim.x) {
        int b   = i / ntot;
        int bin = i - b * ntot;
        float t = (float)tgt[bin] * invHW;
        float c = (float)hin[(size_t)b * MAXBINS + bin] * invHW;
        acc += fabsf(c - t);
    }

    // all 1024 lanes converged here -> EXEC all ones (WMMA requirement)
    v2f a;    a[0] = acc;  a[1] = 0.0f;
    v2f ones; ones[0] = 1.0f; ones[1] = 1.0f;
    v8f c8 = {};
    c8 = __builtin_amdgcn_wmma_f32_16x16x4_f32(
        /*neg_a=*/false, a, /*neg_b=*/false, ones,
        /*c_mod=*/(short)0, c8, /*reuse_a=*/false, /*reuse_b=*/false);
    float half32 = c8[0] + c8[1] + c8[2] + c8[3] + c8[4] + c8[5] + c8[6] + c8[7];
    float wsum = half32 + __shfl_xor(half32, 16, 32);

    const int lane = tid & 31;
    const int wave = tid >> 5;
    if (lane == 0) wsums[wave] = wsum;
    __syncthreads();

    if (tid == 0) {
        float s = 0.0f;
        const int nw = blockDim.x >> 5;
        for (int w = 0; w < nw; ++w) s += wsums[w];
        out[0] = s / (float)total;
    }
}

// ---------------------------------------------------------------------------
extern "C" void kernel_launch(void* const* d_in, const int* in_sizes, int n_in,
                              void* d_out, int out_size, void* d_ws, size_t ws_size,
                              hipStream_t stream) {
    const float* d_input = (const float*)d_in[0];   // [B,3,H,W]
    const float* d_style = (const float*)d_in[1];   // [1,3,H,W]
    const int*   d_nbins = (const int*)d_in[2];     // device scalar
    float*       out     = (float*)d_out;

    const int styleElems = in_sizes[1];             // 3*H*W
    const int HW = styleElems / 3;
    const int B  = in_sizes[0] / styleElems;        // 8

    unsigned* ws_u   = (unsigned*)d_ws;
    unsigned* tgtH   = ws_u;                        // MAXBINS words
    unsigned* inH    = ws_u + MAXBINS;              // B * MAXBINS words

    // 1) zero histogram counters
    const int nwords = (B + 1) * MAXBINS;
    zero_ws_kernel<<<(nwords + 255) / 256, 256, 0, stream>>>(ws_u, nwords);

    // 2) histograms: style (grid.y = 1) and input (grid.y = B)
    const int blocksPerImage = 128;                 // bounds flush traffic
    hist3d_kernel<<<dim3(blocksPerImage, 1), 256, 0, stream>>>(
        d_style, tgtH, d_nbins, HW);
    hist3d_kernel<<<dim3(blocksPerImage, B), 256, 0, stream>>>(
        d_input, inH, d_nbins, HW);

    // 3) L1 mean between normalized histograms (single block, WMMA reduce)
    loss_kernel<<<1, 1024, 0, stream>>>(tgtH, inH, d_nbins, out, B,
                                        1.0f / (float)HW);
}